// BlockGC_67473936221020
// MI455X (gfx1250) — compile-verified
//
#include <hip/hip_runtime.h>
#include <hip/hip_bf16.h>
#include <math.h>
#include <stdint.h>

typedef float v2f __attribute__((ext_vector_type(2)));
typedef float v8f __attribute__((ext_vector_type(8)));
typedef unsigned int u32x4 __attribute__((ext_vector_type(4)));
typedef int i32x8 __attribute__((ext_vector_type(8)));
typedef int i32x4 __attribute__((ext_vector_type(4)));

#define N_      16
#define C_      128
#define T_      128
#define V_      25
#define K_      3
#define H_      8
#define OC_     256
#define HOPMAX_ 12
#define CG_     16                 // C/H
#define TV_     (T_ * V_)          // 3200
#define NTV_    (N_ * TV_)         // 51200 elements per output channel
#define XSLICE_ (C_ * TV_)         // 409600, per-n stride in x
#define OSLICE_ (OC_ * TV_)        // 819200, per-n stride in out/res
#define KK_     (CG_ * V_)         // 400  fused contraction dim
#define NN_     (32 * V_)          // 800  fused output dim per h
#define NNG_    80                 // output cols staged per block (5 wmma tiles)
#define KB_     40                 // k-rows per TDM chunk (10 wmma steps)
#define NCHUNK_ (KK_ / KB_)        // 10

// ---------------------------------------------------------------------------
// TDM helper: async 2D tile load (rows x cols f32, global row stride gstride)
// into LDS at byte offset lds_off.  D# built per CDNA5 ISA 8.3-8.5.
// This toolchain exposes the 6-arg builtin: (g0, g1, g2, g3, g4, cpol).
// ---------------------------------------------------------------------------
__device__ __forceinline__ void tdm_load_2d(const float* gaddr, unsigned lds_off,
                                            int rows, int cols, int gstride) {
    unsigned long long ga = (unsigned long long)(uintptr_t)gaddr;
    u32x4 g0;
    g0.x = 1u;                                        // count=1, user descriptor
    g0.y = lds_off;                                   // lds_addr (bytes)
    g0.z = (unsigned)(ga & 0xFFFFFFFFu);              // global_addr[31:0]
    g0.w = (unsigned)((ga >> 32) & 0x01FFFFFFu)       // global_addr[56:32]
         | (2u << 30);                                // type = 2 ("image")
    i32x8 g1;
    g1[0] = (2 << 16);                                // data_size = 4B
    g1[1] = (gstride & 0xFFFF) << 16;                 // tensor_dim0[15:0]
    g1[2] = ((gstride >> 16) & 0xFFFF)                // tensor_dim0[31:16]
          | ((KK_ & 0xFFFF) << 16);                   // tensor_dim1[15:0]
    g1[3] = ((KK_ >> 16) & 0xFFFF)                    // tensor_dim1[31:16]
          | ((cols & 0xFFFF) << 16);                  // tile_dim0
    g1[4] = (rows & 0xFFFF);                          // tile_dim1 (tile_dim2=0)
    g1[5] = gstride;                                  // tensor_dim0_stride[31:0]
    g1[6] = 0;                                        // stride[47:32], dim1_stride lo
    g1[7] = 0;
    i32x4 gz4 = {0, 0, 0, 0};                         // groups 2/3: dims unused
    i32x8 gz8 = {0, 0, 0, 0, 0, 0, 0, 0};
    __builtin_amdgcn_tensor_load_to_lds(g0, g1, gz4, gz4, gz8, 0);
}

// ---------------------------------------------------------------------------
// 1) BnA[k,h,v',w] = B/||B_col|| + A/||A_col||,  B = emb_table[k,h,hop[v',w]]
// ---------------------------------------------------------------------------
__global__ void prep_bna(const float* __restrict__ emb, const int* __restrict__ hop,
                         const float* __restrict__ A, float* __restrict__ bna) {
    int g = blockIdx.x;            // g = k*H + h
    int j = threadIdx.x;           // column index w
    if (j >= V_) return;
    const float* eg = emb + g * HOPMAX_;
    const float* Ag = A + g * V_ * V_;
    float Bcol[V_];
    float sB = 0.f, sA = 0.f;
#pragma unroll
    for (int i = 0; i < V_; ++i) {
        float b = eg[hop[i * V_ + j]];
        Bcol[i] = b;
        sB += b * b;
        float a = Ag[i * V_ + j];
        sA += a * a;
    }
    float nB = sqrtf(sB) + 1e-4f;
    float nA = sqrtf(sA) + 1e-4f;
    float* og = bna + g * V_ * V_;
#pragma unroll
    for (int i = 0; i < V_; ++i)
        og[i * V_ + j] = Bcol[i] / nB + Ag[i * V_ + j] / nA;
}

// ---------------------------------------------------------------------------
// 2) Fused weight: W2[h][kk=(c'*25+v')][nn=(o'*25+w)]
// ---------------------------------------------------------------------------
__global__ void prep_w2(const float* __restrict__ wb, const float* __restrict__ bna,
                        float* __restrict__ w2) {
    int idx = blockIdx.x * blockDim.x + threadIdx.x;   // < 8*400*800
    int h  = idx / (KK_ * NN_);
    int r  = idx % (KK_ * NN_);
    int kk = r / NN_;
    int nn = r % NN_;
    int cp = kk / V_, vp = kk % V_;
    int op = nn / V_, w  = nn % V_;
    float s = 0.f;
#pragma unroll
    for (int k = 0; k < K_; ++k) {
        int g = k * H_ + h;
        s += wb[(g * 32 + op) * CG_ + cp] * bna[(g * V_ + vp) * V_ + w];
    }
    w2[idx] = s;
}

// ---------------------------------------------------------------------------
// 3) Residual GEMM (WMMA f32 16x16x4).  Each wave: 64 (o) x 16 (m) output,
//    reusing one x-fragment across 4 res_w tiles (4x less x traffic).
// ---------------------------------------------------------------------------
__global__ void __launch_bounds__(256) res_gemm(const float* __restrict__ x,
                                                const float* __restrict__ rw,
                                                float* __restrict__ resp) {
    int wid  = (blockIdx.x * blockDim.x + threadIdx.x) >> 5;
    int lane = threadIdx.x & 31;
    int lo = lane & 15, hi = lane >> 4;

    int n  = wid / 800;            // 800 = 4 o-groups * 200 m-tiles
    int r  = wid % 800;
    int o0 = (r / 200) * 64;
    int m0 = (r % 200) * 16;

    const float* Bcol = x + n * XSLICE_ + m0 + lo;
    const float* A0   = rw + (o0 + lo) * C_;

    v8f acc[4] = {};
    for (int kb = 0; kb < C_; kb += 4) {
        int k0 = kb + 2 * hi;
        v2f b;
        b.x = Bcol[k0 * TV_];
        b.y = Bcol[(k0 + 1) * TV_];
#pragma unroll
        for (int j = 0; j < 4; ++j) {
            v2f a;
            a.x = A0[j * 16 * C_ + k0];
            a.y = A0[j * 16 * C_ + k0 + 1];
            acc[j] = __builtin_amdgcn_wmma_f32_16x16x4_f32(false, a, false, b,
                                                           (short)0, acc[j], false, false);
        }
    }
    float* Out = resp + n * OSLICE_ + m0 + lo;
#pragma unroll
    for (int j = 0; j < 4; ++j)
#pragma unroll
        for (int rr = 0; rr < 8; ++rr)
            Out[(o0 + j * 16 + rr + 8 * hi) * TV_] = acc[j][rr];
}

// ---------------------------------------------------------------------------
// 4) Block-path GEMM per h (M=2048, K=400, N=800), WMMA f32 16x16x4.
//    Block = 8 waves = one full n (128 t-rows) x 80 output cols.
//    W2 B-panel is TDM-loaded into LDS (double buffered, tensorcnt-synced);
//    each wave reuses its A fragment across 5 B tiles.
// ---------------------------------------------------------------------------
__global__ void __launch_bounds__(256) block_gemm(const float* __restrict__ x,
                                                  const float* __restrict__ w2,
                                                  float* __restrict__ outp) {
    __shared__ float panel[2][KB_ * NNG_];             // 2 x 12.8 KB

    int bid = blockIdx.x;                              // h*160 + n*10 + ng
    int h   = bid / 160;
    int r   = bid % 160;
    int n   = r / 10;
    int ng  = r % 10;
    int nn0 = ng * NNG_;

    int wave = threadIdx.x >> 5;
    int lane = threadIdx.x & 31;
    int lo = lane & 15, hi = lane >> 4;

    const float* W2h = w2 + h * (KK_ * NN_);
    int t = wave * 16 + lo;                            // this lane's A row (t index)
    const float* Xbase = x + n * XSLICE_ + (h * CG_) * TV_ + t * V_;

    unsigned lds0 = (unsigned)(uintptr_t)&panel[0][0];
    unsigned lds1 = (unsigned)(uintptr_t)&panel[1][0];

    if (wave == 0)
        tdm_load_2d(W2h + nn0, lds0, KB_, NNG_, NN_);  // chunk 0 -> buf 0

    v8f acc[5] = {};
    for (int c = 0; c < NCHUNK_; ++c) {
        if (wave == 0) {
            if (c + 1 < NCHUNK_) {
                tdm_load_2d(W2h + (c + 1) * KB_ * NN_ + nn0,
                            (c & 1) ? lds0 : lds1, KB_, NNG_, NN_);
                __builtin_amdgcn_s_wait_tensorcnt(1);  // chunk c resident
            } else {
                __builtin_amdgcn_s_wait_tensorcnt(0);
            }
        }
        __syncthreads();                               // panel[c&1] ready for all waves
        const float* P = (c & 1) ? &panel[1][0] : &panel[0][0];
        for (int kk = 0; kk < KB_; kk += 4) {
            int kg = c * KB_ + kk + 2 * hi;            // global k for A gather
            int c0 = kg / V_, v0 = kg % V_;
            int c1 = (kg + 1) / V_, v1 = (kg + 1) % V_;
            v2f a;
            a.x = Xbase[c0 * TV_ + v0];
            a.y = Xbase[c1 * TV_ + v1];
            int kl = kk + 2 * hi;                      // k within chunk for B (LDS)
#pragma unroll
            for (int j = 0; j < 5; ++j) {
                v2f b;
                b.x = P[kl * NNG_ + j * 16 + lo];
                b.y = P[(kl + 1) * NNG_ + j * 16 + lo];
                acc[j] = __builtin_amdgcn_wmma_f32_16x16x4_f32(false, a, false, b,
                                                               (short)0, acc[j], false, false);
            }
        }
        __syncthreads();                               // done reading before reuse
    }

#pragma unroll
    for (int j = 0; j < 5; ++j) {
        int nn = nn0 + j * 16 + lo;
        int op = nn / V_, w = nn % V_;
        float* O = outp + n * OSLICE_ + (h * 32 + op) * TV_ + w;
#pragma unroll
        for (int rr = 0; rr < 8; ++rr) {
            int tt = wave * 16 + rr + 8 * hi;
            O[tt * V_] = acc[j][rr];
        }
    }
}

// ---------------------------------------------------------------------------
// 5) Per-channel mean/var over (n,t,v) for both pre-BN tensors.
// ---------------------------------------------------------------------------
__global__ void bn_stats(const float* __restrict__ outp, const float* __restrict__ resp,
                         float* __restrict__ stats) {
    int o   = blockIdx.x;
    int sel = blockIdx.y;
    const float* buf = sel ? resp : outp;
    float s = 0.f, ss = 0.f;
    for (int j = threadIdx.x; j < NTV_; j += blockDim.x) {
        int n = j / TV_, tv = j % TV_;
        float v = buf[n * OSLICE_ + o * TV_ + tv];
        s += v;
        ss += v * v;
    }
    __shared__ float sh[256], sh2[256];
    sh[threadIdx.x] = s;
    sh2[threadIdx.x] = ss;
    __syncthreads();
    for (int st = 128; st > 0; st >>= 1) {
        if (threadIdx.x < st) {
            sh[threadIdx.x]  += sh[threadIdx.x + st];
            sh2[threadIdx.x] += sh2[threadIdx.x + st];
        }
        __syncthreads();
    }
    if (threadIdx.x == 0) {
        float mean = sh[0] / (float)NTV_;
        float var  = sh2[0] / (float)NTV_ - mean * mean;
        stats[(sel * 256 + o) * 2 + 0] = mean;
        stats[(sel * 256 + o) * 2 + 1] = var;
    }
}

// ---------------------------------------------------------------------------
// 6) out = relu( BN(outp) + BN(resp) ).  (b_block / res_b cancel inside BN.)
// ---------------------------------------------------------------------------
__global__ void finalize(const float* __restrict__ outp, const float* __restrict__ resp,
                         const float* __restrict__ stats,
                         const float* __restrict__ g0, const float* __restrict__ b0,
                         const float* __restrict__ g1, const float* __restrict__ b1,
                         float* __restrict__ out) {
    int idx = blockIdx.x * blockDim.x + threadIdx.x;        // < 13,107,200
    int o = (idx % OSLICE_) / TV_;
    float mo = stats[o * 2],          vo = stats[o * 2 + 1];
    float mr = stats[(256 + o) * 2],  vr = stats[(256 + o) * 2 + 1];
    float a = (outp[idx] - mo) * rsqrtf(vo + 1e-5f) * g0[o] + b0[o];
    float b = (resp[idx] - mr) * rsqrtf(vr + 1e-5f) * g1[o] + b1[o];
    float y = a + b;
    out[idx] = y > 0.f ? y : 0.f;
}

// ---------------------------------------------------------------------------
extern "C" void kernel_launch(void* const* d_in, const int* in_sizes, int n_in,
                              void* d_out, int out_size, void* d_ws, size_t ws_size,
                              hipStream_t stream) {
    const float* x   = (const float*)d_in[0];
    const int*   hop = (const int*)d_in[1];
    const float* emb = (const float*)d_in[2];
    const float* A   = (const float*)d_in[3];
    const float* wb  = (const float*)d_in[4];
    // d_in[5] = b_block  : eliminated by batchnorm mean subtraction (exact)
    const float* bng = (const float*)d_in[6];
    const float* bnb = (const float*)d_in[7];
    const float* rw  = (const float*)d_in[8];
    // d_in[9] = res_b    : eliminated by batchnorm mean subtraction (exact)
    const float* rbg = (const float*)d_in[10];
    const float* rbb = (const float*)d_in[11];
    float* out = (float*)d_out;

    float* ws    = (float*)d_ws;
    float* bna   = ws;                 //    15,000 floats
    float* w2    = ws + 20000;         // 2,560,000 floats
    float* stats = ws + 2600000;       //     1,024 floats
    float* outp  = ws + 2610000;       // 13,107,200 floats
    float* resp  = ws + 15720000;      // 13,107,200 floats (total ~115 MB)

    prep_bna  <<<K_ * H_, 32, 0, stream>>>(emb, hop, A, bna);
    prep_w2   <<<(H_ * KK_ * NN_) / 256, 256, 0, stream>>>(wb, bna, w2);
    res_gemm  <<<1600, 256, 0, stream>>>(x, rw, resp);     // 12800 waves, 64x16/wave
    block_gemm<<<1280, 256, 0, stream>>>(x, w2, outp);     // 8 waves/block, TDM-fed
    bn_stats  <<<dim3(256, 2), 256, 0, stream>>>(outp, resp, stats);
    finalize  <<<(N_ * OSLICE_) / 256, 256, 0, stream>>>(outp, resp, stats,
                                                         bng, bnb, rbg, rbb, out);
}